// Net_86328842649791
// MI455X (gfx1250) — compile-verified
//
#include <hip/hip_runtime.h>
#include <math.h>

#define BATCH 8
#define NPTS  4096
#define KNN   20
#define NBINS 9
#define RAD2DEG 57.29577951308232f

typedef __attribute__((ext_vector_type(2))) float v2f;
typedef __attribute__((ext_vector_type(4))) float v4f;
typedef __attribute__((ext_vector_type(8))) float v8f;

// ---------------------------------------------------------------------------
// Stage 0: transpose src (B,3,N) -> padded points (x,y,z,0) and squared norms
// ---------------------------------------------------------------------------
__global__ __launch_bounds__(256) void prep_kernel(const float* __restrict__ src,
                                                   float* __restrict__ pts4,
                                                   float* __restrict__ xx) {
    int p = blockIdx.x * blockDim.x + threadIdx.x;
    if (p >= BATCH * NPTS) return;
    int b = p >> 12;              // N = 4096
    int n = p & (NPTS - 1);
    const float* s = src + (size_t)b * 3 * NPTS;
    float x = s[n];
    float y = s[NPTS + n];
    float z = s[2 * NPTS + n];
    pts4[(size_t)p * 4 + 0] = x;
    pts4[(size_t)p * 4 + 1] = y;
    pts4[(size_t)p * 4 + 2] = z;
    pts4[(size_t)p * 4 + 3] = 0.0f;
    xx[p] = x * x + y * y + z * z;
}

// monotonic (order-preserving) unsigned mapping of float bits
__device__ __forceinline__ unsigned int fkey32(float f) {
    unsigned int b = __float_as_uint(f);
    unsigned int m = (unsigned int)((int)b >> 31) | 0x80000000u;
    return b ^ m;
}

// ---------------------------------------------------------------------------
// Stage 1: fused WMMA distance tiles + streaming register top-20
// one wave (32 threads) per 16-row tile; 256 column tiles of 16
// A (16x4 f32): lane L -> point tile + (L&15), comps {2*hi, 2*hi+1}, hi = L>>4
// B (4x16 f32): identical per-lane addressing (column point = lane&15)
// C (16x16 f32): vgpr v, lane L -> M = v + 8*hi, N = L&15
// top-k comparator: single u64 key = (fkey(negdist) << 32) | (~idx)
//   -> larger distance first, smaller index wins ties (matches jax top_k)
// LDS tile row stride = 20 floats: 16B-aligned row halves -> ds_load_b128,
// and 20*r mod 64 hits 16 distinct banks across rows.
// ---------------------------------------------------------------------------
#define TSTR 20

__global__ __launch_bounds__(32) void knn_topk_kernel(const float* __restrict__ pts4,
                                                      const float* __restrict__ xx,
                                                      int* __restrict__ idxOut) {
    __shared__ float tile[16 * TSTR];
    __shared__ unsigned long long mk[16 * 40];

    const int b    = blockIdx.x >> 8;      // 256 row tiles per batch
    const int mt   = blockIdx.x & 255;
    const int lane = threadIdx.x;          // 0..31
    const int r    = lane & 15;
    const int hi   = lane >> 4;
    const int base = b * NPTS;

    // A fragment
    const float* ap = pts4 + (size_t)(base + mt * 16 + r) * 4 + 2 * hi;
    v2f afrag;
    afrag.x = ap[0];
    afrag.y = ap[1];

    // row squared-norms for the C-fragment rows this lane will produce
    float xxm[8];
#pragma unroll
    for (int v = 0; v < 8; ++v) xxm[v] = xx[base + mt * 16 + v + 8 * hi];

    // per-lane top-20 as sorted-descending u64 keys (0 = -infinity sentinel)
    unsigned long long key[KNN];
#pragma unroll
    for (int t = 0; t < KNN; ++t) key[t] = 0ull;

    for (int ct = 0; ct < 256; ++ct) {
        const int p = base + ct * 16 + r;
        const float* bp = pts4 + (size_t)p * 4 + 2 * hi;
        v2f bfrag;
        bfrag.x = bp[0];
        bfrag.y = bp[1];

        v8f acc = {};
        acc = __builtin_amdgcn_wmma_f32_16x16x4_f32(
            false, afrag, false, bfrag, (short)0, acc, false, false);

        const float xxn = xx[p];   // column norm for N = lane&15

        __syncthreads();
#pragma unroll
        for (int v = 0; v < 8; ++v) {
            tile[(v + 8 * hi) * TSTR + r] = 2.0f * acc[v] - xxm[v] - xxn;
        }
        __syncthreads();

        // lane L selects for row r over column half [8*hi, 8*hi+8)
        // two 16-byte LDS loads fetch all 8 candidates
        const v4f* tp = (const v4f*)&tile[r * TSTR + 8 * hi];
        v4f c0 = tp[0];
        v4f c1 = tp[1];
        float cand[8];
        cand[0] = c0.x; cand[1] = c0.y; cand[2] = c0.z; cand[3] = c0.w;
        cand[4] = c1.x; cand[5] = c1.y; cand[6] = c1.z; cand[7] = c1.w;

        const int nbase = ct * 16 + 8 * hi;
#pragma unroll
        for (int j = 0; j < 8; ++j) {
            unsigned long long ck =
                ((unsigned long long)fkey32(cand[j]) << 32) |
                (unsigned long long)(0xFFFFFFFFu - (unsigned int)(nbase + j));
            if (ck > key[KNN - 1]) {
#pragma unroll
                for (int t = 0; t < KNN; ++t) {
                    unsigned long long kt = key[t];
                    bool sw = ck > kt;
                    key[t] = sw ? ck : kt;
                    ck     = sw ? kt : ck;
                }
            }
        }
    }

    // merge the two half-lists per row (u64 keys already encode tie-break)
    __syncthreads();
#pragma unroll
    for (int t = 0; t < KNN; ++t) {
        mk[r * 40 + hi * 20 + t] = key[t];
    }
    __syncthreads();
    if (hi == 0) {
        int p0 = 0, p1 = 20;
        const int row = base + mt * 16 + r;
        for (int t = 0; t < KNN; ++t) {
            unsigned long long k0 = mk[r * 40 + p0];
            unsigned long long k1 = mk[r * 40 + p1];
            bool take0 = k0 > k1;
            unsigned long long kk = take0 ? k0 : k1;
            idxOut[(size_t)row * KNN + t] =
                (int)(0xFFFFFFFFu - (unsigned int)(kk & 0xFFFFFFFFull));
            if (take0) ++p0; else ++p1;
        }
    }
}

// ---------------------------------------------------------------------------
// Stage 2: per-point local PCA: top eigenpair of 3x3 scatter of centered knn
// gradient = top eigenvector, magnitude = lambda1^(1/4)
// ---------------------------------------------------------------------------
__global__ __launch_bounds__(256) void eig_kernel(const float* __restrict__ pts4,
                                                  const int* __restrict__ idx,
                                                  float* __restrict__ grads,
                                                  float* __restrict__ mags) {
    int p = blockIdx.x * blockDim.x + threadIdx.x;
    if (p >= BATCH * NPTS) return;
    int b = p >> 12;
    const int* ip = idx + (size_t)p * KNN;
    const int base = b * NPTS;

    float mx = 0.f, my = 0.f, mz = 0.f;
    for (int j = 0; j < KNN; ++j) {
        const float* q = pts4 + (size_t)(base + ip[j]) * 4;
        mx += q[0]; my += q[1]; mz += q[2];
    }
    const float inv_k = 1.0f / (float)KNN;
    mx *= inv_k; my *= inv_k; mz *= inv_k;

    float cxx = 0.f, cxy = 0.f, cxz = 0.f, cyy = 0.f, cyz = 0.f, czz = 0.f;
    for (int j = 0; j < KNN; ++j) {
        const float* q = pts4 + (size_t)(base + ip[j]) * 4;
        float dx = q[0] - mx, dy = q[1] - my, dz = q[2] - mz;
        cxx += dx * dx; cxy += dx * dy; cxz += dx * dz;
        cyy += dy * dy; cyz += dy * dz; czz += dz * dz;
    }

    // largest eigenvalue of symmetric 3x3 (trigonometric closed form)
    float p1 = cxy * cxy + cxz * cxz + cyz * cyz;
    float q  = (cxx + cyy + czz) * (1.0f / 3.0f);
    float b11 = cxx - q, b22 = cyy - q, b33 = czz - q;
    float p2 = b11 * b11 + b22 * b22 + b33 * b33 + 2.0f * p1;
    float lam;
    if (p2 <= 1e-30f) {
        lam = q;
    } else {
        float pp  = sqrtf(p2 * (1.0f / 6.0f));
        float inv = 1.0f / pp;
        float c11 = b11 * inv, c22 = b22 * inv, c33 = b33 * inv;
        float c12 = cxy * inv, c13 = cxz * inv, c23 = cyz * inv;
        float detB = c11 * (c22 * c33 - c23 * c23)
                   - c12 * (c12 * c33 - c23 * c13)
                   + c13 * (c12 * c23 - c22 * c13);
        float rr = fminf(1.0f, fmaxf(-1.0f, 0.5f * detB));
        float phi = acosf(rr) * (1.0f / 3.0f);
        lam = q + 2.0f * pp * cosf(phi);
    }

    // eigenvector via largest cross product of rows of (A - lam*I)
    float r0x = cxx - lam, r0y = cxy,       r0z = cxz;
    float r1x = cxy,       r1y = cyy - lam, r1z = cyz;
    float r2x = cxz,       r2y = cyz,       r2z = czz - lam;

    float ax = r0y * r1z - r0z * r1y, ay = r0z * r1x - r0x * r1z, az = r0x * r1y - r0y * r1x;
    float bx = r0y * r2z - r0z * r2y, by = r0z * r2x - r0x * r2z, bz = r0x * r2y - r0y * r2x;
    float cx = r1y * r2z - r1z * r2y, cy = r1z * r2x - r1x * r2z, cz = r1x * r2y - r1y * r2x;
    float na = ax * ax + ay * ay + az * az;
    float nb = bx * bx + by * by + bz * bz;
    float nc = cx * cx + cy * cy + cz * cz;

    float vx, vy, vz, n2;
    if (na >= nb && na >= nc)      { vx = ax; vy = ay; vz = az; n2 = na; }
    else if (nb >= nc)             { vx = bx; vy = by; vz = bz; n2 = nb; }
    else                           { vx = cx; vy = cy; vz = cz; n2 = nc; }

    if (n2 < 1e-30f) { vx = 0.f; vy = 0.f; vz = 1.f; }
    else {
        float rn = rsqrtf(n2);
        vx *= rn; vy *= rn; vz *= rn;
    }
    // sign convention: largest-|.| component positive
    float fx = fabsf(vx), fy = fabsf(vy), fz = fabsf(vz);
    float msel = (fx >= fy && fx >= fz) ? vx : ((fy >= fz) ? vy : vz);
    if (msel < 0.f) { vx = -vx; vy = -vy; vz = -vz; }

    grads[(size_t)p * 3 + 0] = vx;
    grads[(size_t)p * 3 + 1] = vy;
    grads[(size_t)p * 3 + 2] = vz;
    mags[p] = sqrtf(sqrtf(fmaxf(lam, 0.0f)));
}

// ---------------------------------------------------------------------------
// Stage 3: per-point 9-bin x 2-channel soft histogram + roll + L2 normalize
// ---------------------------------------------------------------------------
__device__ __forceinline__ float pmod180(float x) {
    float r = fmodf(x, 180.0f);
    return (r < 0.0f) ? r + 180.0f : r;
}

__global__ __launch_bounds__(256) void hist_kernel(const int* __restrict__ idx,
                                                   const float* __restrict__ grads,
                                                   const float* __restrict__ mags,
                                                   float* __restrict__ out) {
    int p = blockIdx.x * blockDim.x + threadIdx.x;
    if (p >= BATCH * NPTS) return;
    int b = p >> 12;
    const int base = b * NPTS;
    const int* ip = idx + (size_t)p * KNN;

    float hf[NBINS][2];
    float hs[NBINS][2];
#pragma unroll
    for (int j = 0; j < NBINS; ++j) { hf[j][0] = hf[j][1] = 0.f; hs[j][0] = hs[j][1] = 0.f; }

    for (int j = 0; j < KNN; ++j) {
        int nb = ip[j];
        const float* g = grads + (size_t)(base + nb) * 3;
        float m = mags[base + nb];
        float gz = fminf(1.0f, fmaxf(-1.0f, g[2]));
        float zen = acosf(gz) * RAD2DEG;
        float azi = atanf(g[1] / g[0]) * RAD2DEG;
#pragma unroll
        for (int c = 0; c < 2; ++c) {
            float ang = (float)((int)(c ? azi : zen));  // trunc toward zero
            if (ang < 0.0f) ang += 180.0f;
            int bi = (int)floorf(ang * 0.05f - 0.5f);   // in [-1, 8]
            bi = (bi + NBINS) % NBINS;
            float fc = 20.0f * (float)((bi + 1) % NBINS) + 10.0f;
            float fv = m * pmod180(fc - ang) * 0.05f;
            float sc = 20.0f * (float)bi + 10.0f;
            float sv = m * pmod180(ang - sc) * 0.05f;
#pragma unroll
            for (int t = 0; t < NBINS; ++t) {
                hf[t][c] += (t == bi) ? fv : 0.0f;
                hs[t][c] += (t == bi) ? sv : 0.0f;
            }
        }
    }

    float hist[NBINS][2];
    float nrm[2] = {0.f, 0.f};
#pragma unroll
    for (int t = 0; t < NBINS; ++t) {
#pragma unroll
        for (int c = 0; c < 2; ++c) {
            float h = hf[t][c] + hs[(t + NBINS - 1) % NBINS][c];
            hist[t][c] = h;
            nrm[c] += h * h;
        }
    }
    float in0 = 1.0f / fmaxf(sqrtf(nrm[0]), 1e-12f);
    float in1 = 1.0f / fmaxf(sqrtf(nrm[1]), 1e-12f);
#pragma unroll
    for (int t = 0; t < NBINS; ++t) {
        out[((size_t)p * NBINS + t) * 2 + 0] = hist[t][0] * in0;
        out[((size_t)p * NBINS + t) * 2 + 1] = hist[t][1] * in1;
    }
}

// ---------------------------------------------------------------------------
extern "C" void kernel_launch(void* const* d_in, const int* in_sizes, int n_in,
                              void* d_out, int out_size, void* d_ws, size_t ws_size,
                              hipStream_t stream) {
    const float* src = (const float*)d_in[0];
    // d_in[1] = k (scalar), fixed at compile time

    // workspace layout
    float* pts4  = (float*)d_ws;                               // B*N*4
    float* xx    = pts4 + (size_t)BATCH * NPTS * 4;            // B*N
    float* grads = xx + (size_t)BATCH * NPTS;                  // B*N*3
    float* mags  = grads + (size_t)BATCH * NPTS * 3;           // B*N
    int*   idx   = (int*)(mags + (size_t)BATCH * NPTS);        // B*N*K

    float* out = (float*)d_out;

    const int npt = BATCH * NPTS;
    prep_kernel<<<(npt + 255) / 256, 256, 0, stream>>>(src, pts4, xx);
    knn_topk_kernel<<<BATCH * (NPTS / 16), 32, 0, stream>>>(pts4, xx, idx);
    eig_kernel<<<(npt + 255) / 256, 256, 0, stream>>>(pts4, idx, grads, mags);
    hist_kernel<<<(npt + 255) / 256, 256, 0, stream>>>(idx, grads, mags, out);
}